// Attention_35296041238829
// MI455X (gfx1250) — compile-verified
//
#include <hip/hip_runtime.h>

typedef __attribute__((ext_vector_type(2))) float v2f;
typedef __attribute__((ext_vector_type(8))) float v8f;

#define HIDDEN 256
#define NV_FIX 64

// ---------------------------------------------------------------------------
// Phase 1: dual projection  C = Z*Wc^T + bc,  Y = Z*Wy^T + by
//   Z : [R, 256] row-major, W* : [256(out), 256(hidden)] row-major.
//   One wave32 computes one 16x16 f32 output tile with V_WMMA_F32_16X16X4_F32.
//   A fragment (16x4):  lanes 0-15 K={0,1}, lanes 16-31 K={2,3}, M = lane%16
//   B fragment (4x16):  lanes 0-15 K={0,1}, lanes 16-31 K={2,3}, N = lane%16
//     B[k][n] = W[n][k]  ->  contiguous float2 per lane from W row n.
//   D (16x16):          row = vgpr + 8*(lane/16), col = lane%16
// ---------------------------------------------------------------------------
__global__ __launch_bounds__(256) void proj_wmma_kernel(
    const float* __restrict__ Z,
    const float* __restrict__ Wc, const float* __restrict__ bc,
    const float* __restrict__ Wy, const float* __restrict__ by,
    float* __restrict__ Cout, float* __restrict__ Yout,
    int mTiles, int nTiles)
{
  const int lane = threadIdx.x & 31;
  const int wave = threadIdx.x >> 5;
  const int wg   = blockIdx.x * 8 + wave;        // global wave id == tile id
  const int tilesPerGemm = mTiles * nTiles;

  const int sel   = wg / tilesPerGemm;           // 0 -> (Wc,bc,C), 1 -> (Wy,by,Y)
  int       rem   = wg - sel * tilesPerGemm;
  const int mTile = rem / nTiles;
  const int nTile = rem - mTile * nTiles;

  const float* W    = sel ? Wy   : Wc;
  const float* bias = sel ? by   : bc;
  float*       Out  = sel ? Yout : Cout;

  const int half = lane >> 4;                    // 0: lanes 0-15, 1: lanes 16-31
  const int l    = lane & 15;

  const float* aRow = Z + (size_t)(mTile * 16 + l) * HIDDEN + 2 * half;
  const float* bRow = W + (size_t)(nTile * 16 + l) * HIDDEN + 2 * half;

  v8f acc = {};
#pragma unroll 8
  for (int k = 0; k < HIDDEN; k += 4) {
    v2f a = *(const v2f*)(aRow + k);             // global_load_b64
    v2f b = *(const v2f*)(bRow + k);             // global_load_b64
    // (neg_a, A, neg_b, B, c_mod, C, reuse_a, reuse_b)
    acc = __builtin_amdgcn_wmma_f32_16x16x4_f32(
        false, a, false, b, (short)0, acc, false, false);
  }

  const int   col     = nTile * 16 + l;
  const float bv      = bias[col];
  float*      outBase = Out + (size_t)(mTile * 16 + half * 8) * HIDDEN + col;
#pragma unroll
  for (int v = 0; v < 8; ++v) {
    outBase[(size_t)v * HIDDEN] = acc[v] + bv;
  }
}

// ---------------------------------------------------------------------------
// Phase 2: out[g,i,h] = (1/denom_g) * sum_j sigmoid(c[g,i,h] + y[g,j,h] + mt_ij)
//   mt_ij = (m_i + m_j)*1e10 - 2e10  (exact reference arithmetic; masked pairs
//   drive sigmoid to exactly 0 in f32).
//   One 256-thread block per (g,i); thread = hidden channel; loop j = 0..63.
//   Every y element is consumed exactly once per block -> direct coalesced
//   global loads, L2-resident.
// ---------------------------------------------------------------------------
__global__ __launch_bounds__(256) void pair_sigmoid_kernel(
    const float* __restrict__ Cm, const float* __restrict__ Ym,
    const float* __restrict__ mask, float* __restrict__ out)
{
  const int gi = blockIdx.x;                     // g*NV + i
  const int h  = threadIdx.x;
  const int g  = gi >> 6;                        // NV = 64

  const float ci = Cm[(size_t)gi * HIDDEN + h];
  const float mi = mask[gi];

  const float* yBase = Ym + (size_t)(g * NV_FIX) * HIDDEN + h;
  const float* mBase = mask + g * NV_FIX;

  float acc = 0.0f, denom = 0.0f;
#pragma unroll 4
  for (int j = 0; j < NV_FIX; ++j) {
    const float mj = mBase[j];
    denom += mj;
    const float mt = (mi + mj) * 1.0e10f - 2.0e10f;
    const float x  = ci + yBase[(size_t)j * HIDDEN] + mt;
    acc += 1.0f / (1.0f + __expf(-x));           // sigmoid via v_exp_f32
  }
  out[(size_t)gi * HIDDEN + h] = acc / denom;
}

// ---------------------------------------------------------------------------
// Inputs (setup_inputs order): 0:num_graphs 1:nv 2:z 3:mask 4:Wc 5:bc 6:Wy 7:by
// ---------------------------------------------------------------------------
extern "C" void kernel_launch(void* const* d_in, const int* in_sizes, int n_in,
                              void* d_out, int out_size, void* d_ws, size_t ws_size,
                              hipStream_t stream) {
  (void)n_in; (void)out_size; (void)ws_size;
  const float* z    = (const float*)d_in[2];
  const float* mask = (const float*)d_in[3];
  const float* Wc   = (const float*)d_in[4];
  const float* bc   = (const float*)d_in[5];
  const float* Wy   = (const float*)d_in[6];
  const float* by   = (const float*)d_in[7];
  float*       outp = (float*)d_out;

  const int rows = in_sizes[3];                  // G*NV = 2048
  const int H    = in_sizes[2] / rows;           // 256

  float* Cws = (float*)d_ws;                     // [rows, H]
  float* Yws = Cws + (size_t)rows * H;           // [rows, H]

  const int mTiles = rows / 16;                  // 128
  const int nTiles = H / 16;                     // 16
  const int waves  = mTiles * nTiles * 2;        // 4096 tiles (both GEMMs)

  proj_wmma_kernel<<<waves / 8, 256, 0, stream>>>(z, Wc, bc, Wy, by,
                                                  Cws, Yws, mTiles, nTiles);
  pair_sigmoid_kernel<<<rows, H, 0, stream>>>(Cws, Yws, mask, outp);
}